// NequIP_33543694582314
// MI455X (gfx1250) — compile-verified
//
#include <hip/hip_runtime.h>
#include <hip/hip_bf16.h>
#include <math.h>

#define N_NODESC 10000
#define N_EDGESC 160000
#define D_FEAT   60
#define SCNT     31
#define VCNT     54
#define TCNT     50
#define NIRR     135

typedef float v2f __attribute__((ext_vector_type(2)));
typedef float v8f __attribute__((ext_vector_type(8)));

__device__ __forceinline__ float gelu_f(float x) {
    float x3 = x * x * x;
    return 0.5f * x * (1.0f + tanhf(0.79788456f * (x + 0.044715f * x3)));
}

// M = sum_k a[k] * B_k  (symmetric traceless basis)
__device__ __forceinline__ void mat_from5(const float* a, float M[3][3]) {
    const float r2 = 0.70710678f, r6 = 0.40824829f;
    M[0][1] = M[1][0] = r2 * a[0];
    M[1][2] = M[2][1] = r2 * a[1];
    M[0][2] = M[2][0] = r2 * a[3];
    M[0][0] = -r6 * a[2] + r2 * a[4];
    M[1][1] = -r6 * a[2] - r2 * a[4];
    M[2][2] = 2.0f * r6 * a[2];
}

// to5(sym_tl(M)) == einsum(M, B) directly (B_k symmetric traceless)
__device__ __forceinline__ void to5c(const float M[3][3], float o[5]) {
    const float r2 = 0.70710678f, r6 = 0.40824829f;
    o[0] = r2 * (M[0][1] + M[1][0]);
    o[1] = r2 * (M[1][2] + M[2][1]);
    o[2] = r6 * (2.0f * M[2][2] - M[0][0] - M[1][1]);
    o[3] = r2 * (M[0][2] + M[2][0]);
    o[4] = r2 * (M[0][0] - M[1][1]);
}

// O = cm(v) @ X where cm(v)_{ib} = eps_{iab} v_a
__device__ __forceinline__ void crossmat_mul(const float v[3], const float X[3][3],
                                             float O[3][3]) {
#pragma unroll
    for (int j = 0; j < 3; ++j) {
        O[0][j] = -v[2] * X[1][j] + v[1] * X[2][j];
        O[1][j] =  v[2] * X[0][j] - v[0] * X[2][j];
        O[2][j] = -v[1] * X[0][j] + v[0] * X[1][j];
    }
}

__global__ void copy_kernel(const float* __restrict__ src, float* __restrict__ dst, int n) {
    int i = blockIdx.x * blockDim.x + threadIdx.x;
    if (i < n) dst[i] = src[i];
}

__global__ void zero_kernel(float* __restrict__ dst, int n) {
    int i = blockIdx.x * blockDim.x + threadIdx.x;
    if (i < n) dst[i] = 0.0f;
}

// ---------------------------------------------------------------------------
// Edge kernel: 64 threads = 2 waves; each wave owns 32 edges (two 16-row WMMA
// tiles). Radial MLP runs on V_WMMA_F32_16X16X4_F32; tp + scatter is lane/edge.
// ---------------------------------------------------------------------------
__global__ void __launch_bounds__(64)
edge_kernel(const float* __restrict__ feat,
            const int* __restrict__ senders, const int* __restrict__ receivers,
            const float* __restrict__ Wl_s, const float* __restrict__ Wl_v,
            const float* __restrict__ Wl_t,
            const float* __restrict__ Wr0, const float* __restrict__ Wr1,
            const float* __restrict__ Wr2, const float* __restrict__ Wr3,
            float* __restrict__ aggS, float* __restrict__ aggV,
            float* __restrict__ aggT)
{
    __shared__ float Rst[64][4];        // radial basis staging
    __shared__ float hbuf[2][16][65];   // per-wave hidden 16x64 (padded)
    __shared__ float wbuf[64][137];     // per-edge irrep weights (padded)

    const int tid  = threadIdx.x;
    const int wave = tid >> 5;
    const int lane = tid & 31;
    const int row  = lane & 15;
    const int hi   = lane >> 4;   // half of wave
    const int k0   = hi << 1;     // K sub-offset for f32 wmma frags

    const int e  = blockIdx.x * 64 + tid;   // N_EDGES divisible by 64
    const int si = senders[e];
    const int ri = receivers[e];
    const float* __restrict__ fs = feat + (size_t)si * D_FEAT;
    const float* __restrict__ fr = feat + (size_t)ri * D_FEAT;

    // geometry: r = v_s[0] - v_r[0]
    const float rx = fs[16] - fr[16];
    const float ry = fs[17] - fr[17];
    const float rz = fs[18] - fr[18];
    const float d     = sqrtf(rx * rx + ry * ry + rz * rz);
    const float dsafe = fmaxf(d, 1e-9f);
    const float dinv  = 1.0f / dsafe;
    const float nx = rx * dinv, ny = ry * dinv, nz = rz * dinv;

    // radial basis: sqrt(2/R_MAX)=1, sin(k*pi*d/2)/d
#pragma unroll
    for (int b = 0; b < 4; ++b)
        Rst[tid][b] = sinf((float)(b + 1) * 1.57079632679f * dsafe) * dinv;
    __syncthreads();

    // ================= radial MLP via f32 WMMA =================
    for (int half = 0; half < 2; ++half) {
        const int ebase = wave * 32 + half * 16;
        v8f acc[4];
        {   // layer0: (16x4) x (4x64), scale 1/sqrt(4)=0.5, gelu
            v2f a; a.x = Rst[ebase + row][k0]; a.y = Rst[ebase + row][k0 + 1];
#pragma unroll
            for (int j = 0; j < 4; ++j) {
                const int col = j * 16 + row;
                v2f b; b.x = Wr0[k0 * 64 + col]; b.y = Wr0[(k0 + 1) * 64 + col];
                v8f c = {};
                acc[j] = __builtin_amdgcn_wmma_f32_16x16x4_f32(
                    false, a, false, b, (short)0, c, false, false);
            }
        }
#pragma unroll
        for (int j = 0; j < 4; ++j)
#pragma unroll
            for (int i = 0; i < 8; ++i)
                hbuf[wave][i + 8 * hi][j * 16 + row] = gelu_f(acc[j][i] * 0.5f);
        __syncthreads();

        const float* Wmid[2] = { Wr1, Wr2 };
        for (int layer = 0; layer < 2; ++layer) {   // 64x64, scale 1/8, gelu
            const float* __restrict__ W = Wmid[layer];
            v2f afr[16];
#pragma unroll
            for (int kc = 0; kc < 16; ++kc) {
                afr[kc].x = hbuf[wave][row][4 * kc + k0];
                afr[kc].y = hbuf[wave][row][4 * kc + k0 + 1];
            }
            __syncthreads();
#pragma unroll
            for (int j = 0; j < 4; ++j) {
                const int col = j * 16 + row;
                v8f c = {};
#pragma unroll
                for (int kc = 0; kc < 16; ++kc) {
                    const int kr = 4 * kc + k0;
                    v2f b; b.x = W[kr * 64 + col]; b.y = W[(kr + 1) * 64 + col];
                    c = __builtin_amdgcn_wmma_f32_16x16x4_f32(
                        false, afr[kc], false, b, (short)0, c, false, false);
                }
                acc[j] = c;
            }
#pragma unroll
            for (int j = 0; j < 4; ++j)
#pragma unroll
                for (int i = 0; i < 8; ++i)
                    hbuf[wave][i + 8 * hi][j * 16 + row] = gelu_f(acc[j][i] * 0.125f);
            __syncthreads();
        }

        {   // output layer: 64 -> 135 (9 padded col tiles), scale 1/8, no act
            v2f afr[16];
#pragma unroll
            for (int kc = 0; kc < 16; ++kc) {
                afr[kc].x = hbuf[wave][row][4 * kc + k0];
                afr[kc].y = hbuf[wave][row][4 * kc + k0 + 1];
            }
            for (int j = 0; j < 9; ++j) {
                const int col = j * 16 + row;
                const bool ok = col < NIRR;
                v8f c = {};
#pragma unroll
                for (int kc = 0; kc < 16; ++kc) {
                    const int kr = 4 * kc + k0;
                    v2f b;
                    b.x = ok ? Wr3[kr * NIRR + col] : 0.0f;
                    b.y = ok ? Wr3[(kr + 1) * NIRR + col] : 0.0f;
                    c = __builtin_amdgcn_wmma_f32_16x16x4_f32(
                        false, afr[kc], false, b, (short)0, c, false, false);
                }
                if (ok) {
#pragma unroll
                    for (int i = 0; i < 8; ++i)
                        wbuf[ebase + i + 8 * hi][col] = c[i] * 0.125f;
                }
            }
        }
        __syncthreads();
    }

    // ================= tensor product + weighted scatter =================
    const float r3 = 1.73205081f;
    float a1v[3] = { r3 * nx, r3 * ny, r3 * nz };
    float a2v[5];
    a2v[0] = 3.87298335f * nx * ny;
    a2v[1] = 3.87298335f * ny * nz;
    a2v[2] = 1.11803399f * (3.0f * nz * nz - 1.0f);
    a2v[3] = 3.87298335f * nx * nz;
    a2v[4] = 1.93649167f * (nx * nx - ny * ny);

    float ms[17];
    {
        float sv[16];
#pragma unroll
        for (int m = 0; m < 16; ++m) sv[m] = fs[m];
#pragma unroll
        for (int o = 0; o < 16; ++o) {
            float acc = 0.f;
#pragma unroll
            for (int m = 0; m < 16; ++m) acc += sv[m] * Wl_s[m * 16 + o];
            ms[o] = acc * 0.25f;                 // 1/sqrt(16)
        }
        ms[16] = 1.0f;
    }
    float mv[9][3];
    {
        float vvv[8][3];
#pragma unroll
        for (int m = 0; m < 8; ++m)
#pragma unroll
            for (int c = 0; c < 3; ++c) vvv[m][c] = fs[16 + m * 3 + c];
#pragma unroll
        for (int o = 0; o < 8; ++o)
#pragma unroll
            for (int c = 0; c < 3; ++c) {
                float acc = 0.f;
#pragma unroll
                for (int m = 0; m < 8; ++m) acc += vvv[m][c] * Wl_v[m * 8 + o];
                mv[o][c] = acc * 0.35355339f;    // 1/sqrt(8)
            }
#pragma unroll
        for (int c = 0; c < 3; ++c) mv[8][c] = a1v[c];
    }
    float mt[5][5];
    {
        float tvv[4][5];
#pragma unroll
        for (int m = 0; m < 4; ++m)
#pragma unroll
            for (int c = 0; c < 5; ++c) tvv[m][c] = fs[40 + m * 5 + c];
#pragma unroll
        for (int o = 0; o < 4; ++o)
#pragma unroll
            for (int c = 0; c < 5; ++c) {
                float acc = 0.f;
#pragma unroll
                for (int m = 0; m < 4; ++m) acc += tvv[m][c] * Wl_t[m * 4 + o];
                mt[o][c] = acc * 0.5f;           // 1/sqrt(4)
            }
#pragma unroll
        for (int c = 0; c < 5; ++c) mt[4][c] = a2v[c];
    }

    float A[3][3];
    mat_from5(a2v, A);
    float Tm[5][3][3];
#pragma unroll
    for (int m = 0; m < 5; ++m) mat_from5(mt[m], Tm[m]);
    float P[5][3][3];
#pragma unroll
    for (int m = 0; m < 5; ++m)
#pragma unroll
        for (int i = 0; i < 3; ++i)
#pragma unroll
            for (int j = 0; j < 3; ++j)
                P[m][i][j] = Tm[m][i][0] * A[0][j] + Tm[m][i][1] * A[1][j] +
                             Tm[m][i][2] * A[2][j];

    const float cE = 0.0025f;   // 1/sqrt(N_EDGES)
    const float* __restrict__ w = wbuf[tid];
    float* __restrict__ oS = aggS + (size_t)ri * SCNT;
    float* __restrict__ oV = aggV + (size_t)ri * (VCNT * 3);
    float* __restrict__ oT = aggT + (size_t)ri * (TCNT * 5);

    // ---- scalar irreps (31) ----
#pragma unroll
    for (int c = 0; c < 17; ++c)
        atomicAdd(&oS[c], ms[c] * w[c] * cE);
#pragma unroll
    for (int c = 0; c < 9; ++c) {
        float dot = mv[c][0] * a1v[0] + mv[c][1] * a1v[1] + mv[c][2] * a1v[2];
        atomicAdd(&oS[17 + c], dot * w[17 + c] * cE);
    }
#pragma unroll
    for (int c = 0; c < 5; ++c) {
        float fro = 0.f;
#pragma unroll
        for (int i = 0; i < 3; ++i)
#pragma unroll
            for (int j = 0; j < 3; ++j) fro += Tm[c][i][j] * A[i][j];
        atomicAdd(&oS[26 + c], fro * w[26 + c] * cE);
    }

    // ---- vector irreps (54 x 3) ----
    {
        int vo = 0;
#pragma unroll
        for (int c = 0; c < 17; ++c, ++vo) {
            const float wv = w[SCNT + vo] * cE;
            atomicAdd(&oV[vo * 3 + 0], ms[c] * a1v[0] * wv);
            atomicAdd(&oV[vo * 3 + 1], ms[c] * a1v[1] * wv);
            atomicAdd(&oV[vo * 3 + 2], ms[c] * a1v[2] * wv);
        }
#pragma unroll
        for (int c = 0; c < 9; ++c, ++vo) {
            const float wv = w[SCNT + vo] * cE;
            atomicAdd(&oV[vo * 3 + 0], mv[c][0] * wv);
            atomicAdd(&oV[vo * 3 + 1], mv[c][1] * wv);
            atomicAdd(&oV[vo * 3 + 2], mv[c][2] * wv);
        }
#pragma unroll
        for (int c = 0; c < 9; ++c, ++vo) {   // cross(v, a1)
            const float wv = w[SCNT + vo] * cE;
            atomicAdd(&oV[vo * 3 + 0], (mv[c][1] * a1v[2] - mv[c][2] * a1v[1]) * wv);
            atomicAdd(&oV[vo * 3 + 1], (mv[c][2] * a1v[0] - mv[c][0] * a1v[2]) * wv);
            atomicAdd(&oV[vo * 3 + 2], (mv[c][0] * a1v[1] - mv[c][1] * a1v[0]) * wv);
        }
#pragma unroll
        for (int c = 0; c < 9; ++c, ++vo) {   // A @ v
            const float wv = w[SCNT + vo] * cE;
#pragma unroll
            for (int i = 0; i < 3; ++i)
                atomicAdd(&oV[vo * 3 + i],
                          (A[i][0] * mv[c][0] + A[i][1] * mv[c][1] + A[i][2] * mv[c][2]) * wv);
        }
#pragma unroll
        for (int c = 0; c < 5; ++c, ++vo) {   // Tm @ a1
            const float wv = w[SCNT + vo] * cE;
#pragma unroll
            for (int i = 0; i < 3; ++i)
                atomicAdd(&oV[vo * 3 + i],
                          (Tm[c][i][0] * a1v[0] + Tm[c][i][1] * a1v[1] + Tm[c][i][2] * a1v[2]) * wv);
        }
#pragma unroll
        for (int c = 0; c < 5; ++c, ++vo) {   // eps : P
            const float wv = w[SCNT + vo] * cE;
            atomicAdd(&oV[vo * 3 + 0], (P[c][1][2] - P[c][2][1]) * wv);
            atomicAdd(&oV[vo * 3 + 1], (P[c][2][0] - P[c][0][2]) * wv);
            atomicAdd(&oV[vo * 3 + 2], (P[c][0][1] - P[c][1][0]) * wv);
        }
    }

    // ---- rank-2 irreps (50 x 5) ----
    {
        int to = 0;
#pragma unroll
        for (int c = 0; c < 17; ++c, ++to) {
            const float wt = w[SCNT + VCNT + to] * cE;
#pragma unroll
            for (int k5 = 0; k5 < 5; ++k5)
                atomicAdd(&oT[to * 5 + k5], ms[c] * a2v[k5] * wt);
        }
#pragma unroll
        for (int c = 0; c < 9; ++c, ++to) {   // to5(sym(outer(v, a1)))
            const float wt = w[SCNT + VCNT + to] * cE;
            float M[3][3], q[5];
#pragma unroll
            for (int i = 0; i < 3; ++i)
#pragma unroll
                for (int j = 0; j < 3; ++j) M[i][j] = mv[c][i] * a1v[j];
            to5c(M, q);
#pragma unroll
            for (int k5 = 0; k5 < 5; ++k5)
                atomicAdd(&oT[to * 5 + k5], q[k5] * wt);
        }
#pragma unroll
        for (int c = 0; c < 9; ++c, ++to) {   // to5(sym(cm(v) @ A))
            const float wt = w[SCNT + VCNT + to] * cE;
            float M[3][3], q[5];
            crossmat_mul(mv[c], A, M);
            to5c(M, q);
#pragma unroll
            for (int k5 = 0; k5 < 5; ++k5)
                atomicAdd(&oT[to * 5 + k5], q[k5] * wt);
        }
#pragma unroll
        for (int c = 0; c < 5; ++c, ++to) {   // t * a0
            const float wt = w[SCNT + VCNT + to] * cE;
#pragma unroll
            for (int k5 = 0; k5 < 5; ++k5)
                atomicAdd(&oT[to * 5 + k5], mt[c][k5] * wt);
        }
#pragma unroll
        for (int c = 0; c < 5; ++c, ++to) {   // to5(sym(cm(a1) @ Tm))
            const float wt = w[SCNT + VCNT + to] * cE;
            float M[3][3], q[5];
            crossmat_mul(a1v, Tm[c], M);
            to5c(M, q);
#pragma unroll
            for (int k5 = 0; k5 < 5; ++k5)
                atomicAdd(&oT[to * 5 + k5], q[k5] * wt);
        }
#pragma unroll
        for (int c = 0; c < 5; ++c, ++to) {   // to5(sym(P))
            const float wt = w[SCNT + VCNT + to] * cE;
            float q[5];
            to5c(P[c], q);
#pragma unroll
            for (int k5 = 0; k5 < 5; ++k5)
                atomicAdd(&oT[to * 5 + k5], q[k5] * wt);
        }
    }
}

// ---------------------------------------------------------------------------
// Node update: one lane per node (tiny dense work, L2-resident weights)
// ---------------------------------------------------------------------------
__global__ void __launch_bounds__(64)
node_kernel(float* __restrict__ feat,
            const float* __restrict__ aggS, const float* __restrict__ aggV,
            const float* __restrict__ aggT,
            const float* __restrict__ Wm_s, const float* __restrict__ Wm_v,
            const float* __restrict__ Wm_t,
            const float* __restrict__ Wn_s, const float* __restrict__ Wn_v,
            const float* __restrict__ Wn_t,
            const float* __restrict__ Wp_s, const float* __restrict__ Wp_v,
            const float* __restrict__ Wp_t)
{
    const int n = blockIdx.x * 64 + threadIdx.x;
    if (n >= N_NODESC) return;
    float* f = feat + (size_t)n * D_FEAT;

    float sv[16], vv[8][3], tv[4][5];
#pragma unroll
    for (int m = 0; m < 16; ++m) sv[m] = f[m];
#pragma unroll
    for (int m = 0; m < 8; ++m)
#pragma unroll
        for (int c = 0; c < 3; ++c) vv[m][c] = f[16 + m * 3 + c];
#pragma unroll
    for (int m = 0; m < 4; ++m)
#pragma unroll
        for (int c = 0; c < 5; ++c) tv[m][c] = f[40 + m * 5 + c];

    const float* aS = aggS + (size_t)n * SCNT;
    const float* aV = aggV + (size_t)n * (VCNT * 3);
    const float* aT = aggT + (size_t)n * (TCNT * 5);

    float msv[28];
    for (int o = 0; o < 28; ++o) {
        float a1 = 0.f, a2 = 0.f;
        for (int m = 0; m < SCNT; ++m) a1 += aS[m] * Wm_s[m * 28 + o];
#pragma unroll
        for (int m = 0; m < 16; ++m) a2 += sv[m] * Wn_s[m * 28 + o];
        msv[o] = a1 * 0.17960530f + a2 * 0.25f;      // 1/sqrt(31), 1/sqrt(16)
    }
    float mvv[8][3];
    for (int o = 0; o < 8; ++o)
        for (int c = 0; c < 3; ++c) {
            float a1 = 0.f, a2 = 0.f;
            for (int m = 0; m < VCNT; ++m) a1 += aV[m * 3 + c] * Wm_v[m * 8 + o];
#pragma unroll
            for (int m = 0; m < 8; ++m) a2 += vv[m][c] * Wn_v[m * 8 + o];
            mvv[o][c] = a1 * 0.13608276f + a2 * 0.35355339f;  // 1/sqrt(54), 1/sqrt(8)
        }
    float mtv[4][5];
    for (int o = 0; o < 4; ++o)
        for (int c = 0; c < 5; ++c) {
            float a1 = 0.f, a2 = 0.f;
            for (int m = 0; m < TCNT; ++m) a1 += aT[m * 5 + c] * Wm_t[m * 4 + o];
#pragma unroll
            for (int m = 0; m < 4; ++m) a2 += tv[m][c] * Wn_t[m * 4 + o];
            mtv[o][c] = a1 * 0.14142136f + a2 * 0.5f;         // 1/sqrt(50), 1/sqrt(4)
        }

    float g[12];
#pragma unroll
    for (int j = 0; j < 12; ++j) g[j] = 1.0f / (1.0f + expf(-msv[16 + j]));
    float gs[16];
#pragma unroll
    for (int m = 0; m < 16; ++m) gs[m] = gelu_f(msv[m]);

#pragma unroll
    for (int c = 0; c < 16; ++c) {
        float acc = 0.f;
#pragma unroll
        for (int m = 0; m < 16; ++m) acc += gs[m] * Wp_s[m * 16 + c];
        f[c] = acc * 0.25f;
    }
#pragma unroll
    for (int o = 0; o < 8; ++o)
#pragma unroll
        for (int c = 0; c < 3; ++c) {
            float acc = 0.f;
#pragma unroll
            for (int m = 0; m < 8; ++m) acc += mvv[m][c] * g[m] * Wp_v[m * 8 + o];
            f[16 + o * 3 + c] = acc * 0.35355339f;
        }
#pragma unroll
    for (int o = 0; o < 4; ++o)
#pragma unroll
        for (int c = 0; c < 5; ++c) {
            float acc = 0.f;
#pragma unroll
            for (int m = 0; m < 4; ++m) acc += mtv[m][c] * g[8 + m] * Wp_t[m * 4 + o];
            f[40 + o * 5 + c] = acc * 0.5f;
        }
}

extern "C" void kernel_launch(void* const* d_in, const int* in_sizes, int n_in,
                              void* d_out, int out_size, void* d_ws, size_t ws_size,
                              hipStream_t stream) {
    (void)in_sizes; (void)n_in; (void)out_size; (void)ws_size;
    const float* x    = (const float*)d_in[0];
    const float* Wl_s = (const float*)d_in[1];
    const float* Wl_v = (const float*)d_in[2];
    const float* Wl_t = (const float*)d_in[3];
    const float* Wr0  = (const float*)d_in[4];
    const float* Wr1  = (const float*)d_in[5];
    const float* Wr2  = (const float*)d_in[6];
    const float* Wr3  = (const float*)d_in[7];
    const float* Wm_s = (const float*)d_in[8];
    const float* Wm_v = (const float*)d_in[9];
    const float* Wm_t = (const float*)d_in[10];
    const float* Wn_s = (const float*)d_in[11];
    const float* Wn_v = (const float*)d_in[12];
    const float* Wn_t = (const float*)d_in[13];
    const float* Wp_s = (const float*)d_in[14];
    const float* Wp_v = (const float*)d_in[15];
    const float* Wp_t = (const float*)d_in[16];
    const int* senders   = (const int*)d_in[17];
    const int* receivers = (const int*)d_in[18];

    float* feat = (float*)d_ws;
    float* aggS = feat + (size_t)N_NODESC * D_FEAT;
    float* aggV = aggS + (size_t)N_NODESC * SCNT;
    float* aggT = aggV + (size_t)N_NODESC * VCNT * 3;

    const int nf = N_NODESC * D_FEAT;
    copy_kernel<<<(nf + 255) / 256, 256, 0, stream>>>(x, feat, nf);

    const int nagg = N_NODESC * (SCNT + VCNT * 3 + TCNT * 5);
    for (int k = 0; k < 3; ++k) {
        zero_kernel<<<(nagg + 255) / 256, 256, 0, stream>>>(aggS, nagg);
        edge_kernel<<<N_EDGESC / 64, 64, 0, stream>>>(
            feat, senders, receivers,
            Wl_s + k * 16 * 16, Wl_v + k * 8 * 8, Wl_t + k * 4 * 4,
            Wr0 + k * 4 * 64, Wr1 + k * 64 * 64, Wr2 + k * 64 * 64,
            Wr3 + k * 64 * NIRR,
            aggS, aggV, aggT);
        node_kernel<<<(N_NODESC + 63) / 64, 64, 0, stream>>>(
            feat, aggS, aggV, aggT,
            Wm_s + k * SCNT * 28, Wm_v + k * VCNT * 8, Wm_t + k * TCNT * 4,
            Wn_s + k * 16 * 28, Wn_v + k * 8 * 8, Wn_t + k * 4 * 4,
            Wp_s + k * 16 * 16, Wp_v + k * 8 * 8, Wp_t + k * 4 * 4);
    }
    copy_kernel<<<(nf + 255) / 256, 256, 0, stream>>>(feat, (float*)d_out, nf);
}